// BoothGroupQuant_67465346285677
// MI455X (gfx1250) — compile-verified
//
#include <hip/hip_runtime.h>
#include <stdint.h>

// Booth/NAF group quantizer for MI455X (gfx1250, wave32).
// Layout: 1 thread = 1 group of 16 elements (64 B). Tiles of 256 threads
// (16 KB) are staged through LDS with the CDNA5 async global<->LDS engine so
// all HBM traffic is lane-linear b128 (coalesced), while the 64B-strided
// per-thread gathers hit LDS instead of global.

#define SF        0.0078125f
#define SF_INV    128.0f
#define GROUP     16
#define KMAX      8
#define TPB       256
#define TILE_ELEMS (TPB * GROUP)   // 4096 floats = 16 KB per buffer

static __device__ __forceinline__ uint32_t lds_off(const void* p) {
  // Generic pointers to LDS carry the shared aperture in the high 32 bits;
  // low 32 bits are the workgroup-relative LDS byte offset (ISA 10.2).
  return (uint32_t)(uintptr_t)p;
}

static __device__ __forceinline__ void async_load16(uint32_t lds_byte_off, const void* gaddr) {
  // VDST = LDS byte address, VADDR = 64-bit global address, saddr off.
  asm volatile("global_load_async_to_lds_b128 %0, %1, off"
               :: "v"(lds_byte_off), "v"(gaddr) : "memory");
}

static __device__ __forceinline__ void async_store16(void* gaddr, uint32_t lds_byte_off) {
  // VADDR = 64-bit global address, VSRC = LDS byte address.
  asm volatile("global_store_async_from_lds_b128 %0, %1, off"
               :: "v"(gaddr), "v"(lds_byte_off) : "memory");
}

static __device__ __forceinline__ void wait_async0() {
  asm volatile("s_wait_asynccnt 0" ::: "memory");
}

// One group of 16 elements: NAF decompose, keep the 8 largest-exponent terms
// (ties -> lower element index), reconstruct.
static __device__ __forceinline__ void booth_group16(const float* xi, float* ov) {
  uint32_t pp[GROUP];  // +1 digit position masks
  uint32_t um[GROUP];  // all nonzero digit position masks
  int total = 0;
#pragma unroll
  for (int i = 0; i < GROUP; ++i) {
    float rf = rintf(xi[i] * SF_INV);                 // round-half-even like jnp.round
    rf = fminf(fmaxf(rf, -32768.0f), 32767.0f);       // clip after rounding
    int v = (int)rf;
    int t = 3 * v;
    uint32_t plus  = ((uint32_t)(t & ~v)) >> 1;       // NAF +1 digits (bit p+1 of 3v&~v)
    uint32_t minus = ((uint32_t)(v & ~t)) >> 1;       // NAF -1 digits
    pp[i] = plus;
    um[i] = plus | minus;
    total += __popc(plus | minus);
  }

  int estar = 0, rleft = 0;
  uint32_t gt = 0xFFFFFFFFu;                          // keep-all when total <= 8
  if (total > KMAX) {
    uint32_t orall = 0;
#pragma unroll
    for (int i = 0; i < GROUP; ++i) orall |= um[i];
    int kept = 0;
    int e = 31 - __clz((int)orall);                   // highest occupied exponent
    for (;; --e) {
      int c = 0;
#pragma unroll
      for (int i = 0; i < GROUP; ++i) c += (int)((um[i] >> e) & 1u);
      if (kept + c >= KMAX) break;                    // threshold exponent found
      kept += c;
    }
    estar = e;
    rleft = KMAX - kept;                              // budget left at e*
    gt = ~((2u << estar) - 1u);                       // keep every digit with exp > e*
  }

  int cnt = 0;
#pragma unroll
  for (int i = 0; i < GROUP; ++i) {
    uint32_t has = (um[i] >> estar) & 1u;
    uint32_t k = gt;
    if (has && cnt < rleft) k |= (1u << estar);       // first rleft elements win ties
    int val = (int)(pp[i] & k) - (int)((um[i] ^ pp[i]) & k);  // um^pp = minus mask
    ov[i] = (float)val * SF;
    cnt += (int)has;
  }
}

__global__ __launch_bounds__(TPB)
void BoothGroupQuant_kernel(const float* __restrict__ x,
                            float* __restrict__ out,
                            long long n) {
  __shared__ float sIn[TILE_ELEMS];
  __shared__ float sOut[TILE_ELEMS];

  const int t = threadIdx.x;
  const long long tileBase = (long long)blockIdx.x * TILE_ELEMS;

  if (tileBase + TILE_ELEMS <= n) {
    // ---- Stage 16 KB tile into LDS with coalesced async b128 loads ----
    const char* gsrc = (const char*)(x + tileBase);
    const uint32_t inBase = lds_off(sIn);
#pragma unroll
    for (int k = 0; k < 4; ++k) {
      uint32_t boff = (uint32_t)((k * TPB + t) * 16);
      async_load16(inBase + boff, gsrc + boff);
    }
    wait_async0();        // own wave's ASYNCcnt
    __syncthreads();      // cross-wave visibility of LDS tile

    // ---- Compute: each thread consumes its 64 B group from LDS ----
    float xi[GROUP], ov[GROUP];
    const float4* s4 = (const float4*)(sIn + t * GROUP);
#pragma unroll
    for (int k = 0; k < 4; ++k) {
      float4 v = s4[k];
      xi[4 * k + 0] = v.x; xi[4 * k + 1] = v.y;
      xi[4 * k + 2] = v.z; xi[4 * k + 3] = v.w;
    }
    booth_group16(xi, ov);
    float4* d4 = (float4*)(sOut + t * GROUP);
#pragma unroll
    for (int k = 0; k < 4; ++k) {
      float4 v;
      v.x = ov[4 * k + 0]; v.y = ov[4 * k + 1];
      v.z = ov[4 * k + 2]; v.w = ov[4 * k + 3];
      d4[k] = v;
    }
    __syncthreads();

    // ---- Drain results with coalesced async b128 stores from LDS ----
    char* gdst = (char*)(out + tileBase);
    const uint32_t outBase = lds_off(sOut);
#pragma unroll
    for (int k = 0; k < 4; ++k) {
      uint32_t boff = (uint32_t)((k * TPB + t) * 16);
      async_store16(gdst + boff, outBase + boff);
    }
    wait_async0();
  } else {
    // Partial tile fallback: direct global path, per-group bounds check.
    const long long g = tileBase + (long long)t * GROUP;
    if (g + GROUP <= n) {
      float xi[GROUP], ov[GROUP];
#pragma unroll
      for (int i = 0; i < GROUP; ++i) xi[i] = x[g + i];
      booth_group16(xi, ov);
#pragma unroll
      for (int i = 0; i < GROUP; ++i) out[g + i] = ov[i];
    }
  }
}

extern "C" void kernel_launch(void* const* d_in, const int* in_sizes, int n_in,
                              void* d_out, int out_size, void* d_ws, size_t ws_size,
                              hipStream_t stream) {
  const float* x = (const float*)d_in[0];
  float* out = (float*)d_out;
  const long long n = (long long)in_sizes[0];
  const long long groups = n / GROUP;
  const int blocks = (int)((groups + TPB - 1) / TPB);
  BoothGroupQuant_kernel<<<blocks, TPB, 0, stream>>>(x, out, n);
}